// GNNPolicyExtractor_65197603553736
// MI455X (gfx1250) — compile-verified
//
#include <hip/hip_runtime.h>

typedef __attribute__((ext_vector_type(2))) float v2f;
typedef __attribute__((ext_vector_type(8))) float v8f;

#define N_NODES_C 100000
#define N_EDGES_C 3200000
#define IN_CH_C   128
#define HID_C     64

// ---------------- utility ----------------
__global__ void zero_kernel(float* __restrict__ p, int n) {
  int i = blockIdx.x * blockDim.x + threadIdx.x;
  if (i < n) p[i] = 0.0f;
}

// deg[dst] += 1 (self-loop added later in rsqrt kernel)
__global__ void degree_kernel(const int* __restrict__ dst, float* __restrict__ deg, int n_edges) {
  int e = blockIdx.x * blockDim.x + threadIdx.x;
  if (e < n_edges) atomicAdd(&deg[dst[e]], 1.0f);
}

__global__ void rsqrt_kernel(float* __restrict__ deg, int n) {
  int i = blockIdx.x * blockDim.x + threadIdx.x;
  if (i < n) deg[i] = rsqrtf(deg[i] + 1.0f);
}

// ---------------- WMMA GEMM: H[n_rows x 64] = A[n_rows x K] * W[K x 64] ----------------
// One wave computes one 16x16 output tile using V_WMMA_F32_16X16X4_F32.
// Fragment layouts per CDNA5 ISA (wave32):
//   A 16x4 : lane%16 = M row; VGPR v holds K = v + 2*(lane>=16)
//   B 4x16 : lane%16 = N col; VGPR v holds K = v + 2*(lane>=16)
//   C 16x16: VGPR r holds M = r + 8*(lane>=16), N = lane%16
__global__ void gemm_wmma_kernel(const float* __restrict__ A, const float* __restrict__ W,
                                 float* __restrict__ H, int n_rows, int K) {
  __shared__ float sW[IN_CH_C * HID_C];  // up to 128x64 f32 = 32 KB
  const int nW = K * HID_C;
  for (int i = threadIdx.x; i < nW; i += blockDim.x) sW[i] = W[i];
  __syncthreads();

  const int gwave  = (blockIdx.x * blockDim.x + threadIdx.x) >> 5;
  const int lane   = threadIdx.x & 31;
  const int tile_m = gwave >> 2;       // 64/16 = 4 column tiles
  const int tile_n = gwave & 3;
  if (tile_m * 16 >= n_rows) return;   // uniform per wave (never hit for N=100000)

  const int rc    = lane & 15;             // row (A) / col (B,C)
  const int khalf = (lane >> 4) << 1;      // 0 or 2

  const float* __restrict__ arow = A + (size_t)(tile_m * 16 + rc) * K + khalf;
  const float* __restrict__ bcol = sW + khalf * HID_C + (tile_n * 16 + rc);

  v8f c = {};
  for (int kk = 0; kk < K; kk += 4) {
    v2f a, b;
    a.x = arow[kk];
    a.y = arow[kk + 1];
    b.x = bcol[kk * HID_C];
    b.y = bcol[kk * HID_C + HID_C];
    c = __builtin_amdgcn_wmma_f32_16x16x4_f32(
        /*neg_a=*/false, a, /*neg_b=*/false, b,
        /*c_mod=*/(short)0, c, /*reuse_a=*/false, /*reuse_b=*/false);
  }

  const int m0 = tile_m * 16 + ((lane >> 4) << 3);
  float* __restrict__ out = H + (size_t)m0 * HID_C + tile_n * 16 + rc;
#pragma unroll
  for (int r = 0; r < 8; ++r) out[r * HID_C] = c[r];
}

// ---------------- edge scatter: agg[dst] += h[src] * dis[src]*dis[dst] ----------------
// one thread = one edge x 4 channels (float4 gather, 4 f32 atomics; agg fits in L2)
__global__ void scatter_kernel(const float* __restrict__ h, const int* __restrict__ src,
                               const int* __restrict__ dst, const float* __restrict__ dis,
                               float* __restrict__ agg, int n_edges) {
  unsigned idx = blockIdx.x * blockDim.x + threadIdx.x;
  unsigned e = idx >> 4;
  if (e >= (unsigned)n_edges) return;
  const int c = (int)(idx & 15u) << 2;
  const int s = src[e];
  const int d = dst[e];
  const float norm = dis[s] * dis[d];
  const float4 hv = *(const float4*)(h + (size_t)s * HID_C + c);
  float* __restrict__ ap = agg + (size_t)d * HID_C + c;
  atomicAdd(ap + 0, hv.x * norm);
  atomicAdd(ap + 1, hv.y * norm);
  atomicAdd(ap + 2, hv.z * norm);
  atomicAdd(ap + 3, hv.w * norm);
}

// ---------------- agg = relu(agg + h*dis^2 + b) in place ----------------
__global__ void finalize_relu_kernel(float* __restrict__ agg, const float* __restrict__ h,
                                     const float* __restrict__ dis, const float* __restrict__ bias,
                                     int n) {
  int idx = blockIdx.x * blockDim.x + threadIdx.x;
  if (idx >= n) return;
  const int node = idx >> 6;
  const int c = idx & 63;
  const float di = dis[node];
  const float v = agg[idx] + h[idx] * di * di + bias[c];
  agg[idx] = fmaxf(v, 0.0f);
}

// ---------------- layer-2 finalize fused with column-sum reduction ----------------
__global__ void finalize_reduce_kernel(const float* __restrict__ agg, const float* __restrict__ h,
                                       const float* __restrict__ dis, const float* __restrict__ bias,
                                       float* __restrict__ acc, int n) {
  __shared__ float s[HID_C];
  const int t = threadIdx.x;
  if (t < HID_C) s[t] = 0.0f;
  __syncthreads();
  const int idx = blockIdx.x * blockDim.x + t;
  if (idx < n) {
    const int node = idx >> 6;
    const int c = idx & 63;
    const float di = dis[node];
    const float v = fmaxf(agg[idx] + h[idx] * di * di + bias[c], 0.0f);
    atomicAdd(&s[c], v);  // LDS f32 atomic (ds_add_f32)
  }
  __syncthreads();
  if (t < HID_C) atomicAdd(&acc[t], s[t]);
}

__global__ void mean_out_kernel(const float* __restrict__ acc, float* __restrict__ out) {
  const int c = threadIdx.x;
  if (c < HID_C) out[c] = acc[c] * (1.0f / (float)N_NODES_C);
}

// ---------------- host-side launch ----------------
extern "C" void kernel_launch(void* const* d_in, const int* in_sizes, int n_in,
                              void* d_out, int out_size, void* d_ws, size_t ws_size,
                              hipStream_t stream) {
  const float* x  = (const float*)d_in[0];
  const int*   ei = (const int*)d_in[1];   // edge_index [2, E] (int)
  const float* W1 = (const float*)d_in[2];
  const float* b1 = (const float*)d_in[3];
  const float* W2 = (const float*)d_in[4];
  const float* b2 = (const float*)d_in[5];
  float* out = (float*)d_out;

  const int* src = ei;
  const int* dst = ei + N_EDGES_C;

  // workspace layout (all f32): dis[N] | bufA[N*64] | bufB[N*64] | acc[64]  (~51.6 MB)
  float* dis  = (float*)d_ws;
  float* bufA = dis + N_NODES_C;
  float* bufB = bufA + (size_t)N_NODES_C * HID_C;
  float* acc  = bufB + (size_t)N_NODES_C * HID_C;

  const int NH = N_NODES_C * HID_C;                 // 6.4M
  const dim3 blk(256);
  const int g_nodes = (N_NODES_C + 255) / 256;
  const int g_edges = (N_EDGES_C + 255) / 256;
  const int g_nh    = (NH + 255) / 256;
  const unsigned g_scatter = ((unsigned)N_EDGES_C * 16u + 255u) / 256u;
  const int tiles   = ((N_NODES_C + 15) / 16) * (HID_C / 16);   // 25000 waves
  const int g_gemm  = (tiles * 32 + 255) / 256;                 // 3125 blocks

  // degree + normalization
  zero_kernel<<<g_nodes, blk, 0, stream>>>(dis, N_NODES_C);
  zero_kernel<<<1, 64, 0, stream>>>(acc, HID_C);
  degree_kernel<<<g_edges, blk, 0, stream>>>(dst, dis, N_EDGES_C);
  rsqrt_kernel<<<g_nodes, blk, 0, stream>>>(dis, N_NODES_C);

  // layer 1
  gemm_wmma_kernel<<<g_gemm, blk, 0, stream>>>(x, W1, bufA, N_NODES_C, IN_CH_C);
  zero_kernel<<<g_nh, blk, 0, stream>>>(bufB, NH);
  scatter_kernel<<<g_scatter, blk, 0, stream>>>(bufA, src, dst, dis, bufB, N_EDGES_C);
  finalize_relu_kernel<<<g_nh, blk, 0, stream>>>(bufB, bufA, dis, b1, NH);

  // layer 2 (bufB = y1 is GEMM input; stream order makes the re-zero safe)
  gemm_wmma_kernel<<<g_gemm, blk, 0, stream>>>(bufB, W2, bufA, N_NODES_C, HID_C);
  zero_kernel<<<g_nh, blk, 0, stream>>>(bufB, NH);
  scatter_kernel<<<g_scatter, blk, 0, stream>>>(bufA, src, dst, dis, bufB, N_EDGES_C);
  finalize_reduce_kernel<<<g_nh, blk, 0, stream>>>(bufB, bufA, dis, b2, acc, NH);
  mean_out_kernel<<<1, 64, 0, stream>>>(acc, out);
}